// KMeansBase_24043226923147
// MI455X (gfx1250) — compile-verified
//
#include <hip/hip_runtime.h>
#include <math.h>
#include <stdint.h>

#define BATCH   8
#define NPTS    65536
#define FDIM    64
#define KCLUST  32
#define SEED_BLOCKS 64   // blocks per batch for each seeding pass

typedef float v2f __attribute__((ext_vector_type(2)));
typedef float v8f __attribute__((ext_vector_type(8)));

// D(16x16,f32) = A(16x4,f32) x B(4x16,f32) + C ; wave32, striped across lanes.
__device__ inline v8f wmma_f32_16x16x4(v2f a, v2f b, v8f c) {
#if defined(__HIP_DEVICE_COMPILE__)
  return __builtin_amdgcn_wmma_f32_16x16x4_f32(
      /*neg_a=*/false, a, /*neg_b=*/false, b,
      /*c_mod=*/(short)0, c, /*reuse_a=*/false, /*reuse_b=*/false);
#else
  (void)a; (void)b;
  return c;
#endif
}

__device__ inline unsigned wang_hash(unsigned s) {
  s = (s ^ 61u) ^ (s >> 16);
  s *= 9u;
  s = s ^ (s >> 4);
  s *= 0x27d4eb2du;
  s = s ^ (s >> 15);
  return s;
}

// ---------------------------------------------------------------------------
// Seeding, device-wide. Pass k: update min d^2 against centroid k-1 and emit
// a per-block argmax; a small per-batch kernel reduces the partials and
// gathers centroid k. 512 blocks per pass -> data stays L2-resident and every
// WGP participates (vs. 8 WGPs in a one-block-per-batch design).
// ---------------------------------------------------------------------------
__global__ void seed_first_kernel(const float* __restrict__ data,
                                  float* __restrict__ cent) {
  const int b   = blockIdx.x;
  const int tid = threadIdx.x;  // 64 == FDIM
  const unsigned pick = wang_hash((unsigned)b * 2654435761u + 42u) % NPTS;
  cent[((size_t)b * KCLUST + 0) * FDIM + tid] =
      data[((size_t)b * NPTS + pick) * FDIM + tid];
}

__global__ __launch_bounds__(256) void seed_update_kernel(
    const float* __restrict__ data, const float* __restrict__ cent,
    float* __restrict__ min_d, float* __restrict__ partV,
    int* __restrict__ partI, int k, int first) {
  const int b   = blockIdx.y;
  const int tid = threadIdx.x;
  __shared__ float sC[FDIM];
  __shared__ float sV[256];
  __shared__ int   sI[256];

  if (tid < FDIM) sC[tid] = cent[((size_t)b * KCLUST + (k - 1)) * FDIM + tid];
  __syncthreads();

  const float* __restrict__ X  = data + (size_t)b * NPTS * FDIM;
  float* __restrict__       md = min_d + (size_t)b * NPTS;

  float best = -1.f;
  int   bi   = 0;
  for (int n = blockIdx.x * 256 + tid; n < NPTS; n += gridDim.x * 256) {
    const float* row = X + (size_t)n * FDIM;
    float d2 = 0.f;
#pragma unroll
    for (int f = 0; f < FDIM; ++f) {
      float df = row[f] - sC[f];
      d2 = fmaf(df, df, d2);
    }
    if (!first) {
      float m = md[n];
      d2 = d2 < m ? d2 : m;
    }
    md[n] = d2;
    if (d2 > best) { best = d2; bi = n; }
  }
  sV[tid] = best;
  sI[tid] = bi;
  __syncthreads();
  for (int off = 128; off > 0; off >>= 1) {
    if (tid < off && sV[tid + off] > sV[tid]) {
      sV[tid] = sV[tid + off];
      sI[tid] = sI[tid + off];
    }
    __syncthreads();
  }
  if (tid == 0) {
    partV[b * gridDim.x + blockIdx.x] = sV[0];
    partI[b * gridDim.x + blockIdx.x] = sI[0];
  }
}

__global__ void seed_select_kernel(const float* __restrict__ data,
                                   const float* __restrict__ partV,
                                   const int* __restrict__ partI,
                                   float* __restrict__ cent, int k) {
  const int b   = blockIdx.x;
  const int tid = threadIdx.x;  // SEED_BLOCKS == 64 == FDIM
  __shared__ float sV[SEED_BLOCKS];
  __shared__ int   sI[SEED_BLOCKS];
  __shared__ int   sPick;
  sV[tid] = partV[b * SEED_BLOCKS + tid];
  sI[tid] = partI[b * SEED_BLOCKS + tid];
  __syncthreads();
  for (int off = SEED_BLOCKS / 2; off > 0; off >>= 1) {
    if (tid < off && sV[tid + off] > sV[tid]) {
      sV[tid] = sV[tid + off];
      sI[tid] = sI[tid + off];
    }
    __syncthreads();
  }
  if (tid == 0) sPick = sI[0];
  __syncthreads();
  const int pick = sPick;
  cent[((size_t)b * KCLUST + k) * FDIM + tid] =
      data[((size_t)b * NPTS + pick) * FDIM + tid];
}

// ---------------------------------------------------------------------------
// Distance + argmin. 256 threads = 8 wave32; each wave owns a 16-row strip.
// Centroid tile (8 KB) staged into LDS with gfx1250 async global->LDS copies
// (ASYNCcnt-tracked), ||c||^2 computed in-LDS, dot products via f32 WMMA
// 16x16x4 over F=64 (16 steps) x two 16-wide K tiles.
//
// A fragment (16x4 f32): lanes 0-15 hold M=lane, K={0,1}; lanes 16-31 hold
// M=lane-16, K={2,3}. Step s covers features 4s..4s+3 -> lane (h,l16)
// supplies data[row0+l16][4s+2h .. +1], one contiguous 8-byte load.
// C/D layout: acc[i] on lane = element (M = i + 8h, N = l16).
// ---------------------------------------------------------------------------
__global__ __launch_bounds__(256) void kmeans_dist_kernel(
    const float* __restrict__ data, const float* __restrict__ cent,
    float* __restrict__ dist, int* __restrict__ ids) {
  __shared__ float sC[KCLUST * FDIM];      // 8 KB centroids
  __shared__ float sCn[KCLUST];
  __shared__ float sP[8][32];              // per-wave row-norm partials
  __shared__ float sD[8][16 * KCLUST];     // per-wave distance tile (16 KB)

  const int b   = blockIdx.y;
  const int tid = threadIdx.x;

  // Async-stage centroids: 2048 floats = 512 x B128, 2 per thread.
  {
    const float* __restrict__ gsrc = cent + (size_t)b * KCLUST * FDIM;
#pragma unroll
    for (int it = 0; it < (KCLUST * FDIM) / (256 * 4); ++it) {
      const int i = (it * 256 + tid) * 4;
      unsigned lds_off = (unsigned)(uintptr_t)(&sC[i]);
      const float* g = gsrc + i;
      asm volatile("global_load_async_to_lds_b128 %0, %1, off"
                   :: "v"(lds_off), "v"(g) : "memory");
    }
    asm volatile("s_wait_asynccnt 0" ::: "memory");
  }
  __syncthreads();

  if (tid < KCLUST) {  // ||c||^2 from the LDS-resident tile
    const float* c = sC + tid * FDIM;
    float s = 0.f;
#pragma unroll
    for (int f = 0; f < FDIM; ++f) s = fmaf(c[f], c[f], s);
    sCn[tid] = s;
  }
  __syncthreads();

  const int wave = tid >> 5;
  const int lane = tid & 31;
  const int h    = lane >> 4;
  const int l16  = lane & 15;
  const int row0 = blockIdx.x * 128 + wave * 16;

  const v2f* __restrict__ Xp =
      (const v2f*)(data + ((size_t)b * NPTS + row0 + l16) * FDIM);

  v2f a[16];
  float pn = 0.f;  // partial ||x||^2 for row l16 (features f%4 in {2h,2h+1})
#pragma unroll
  for (int s = 0; s < 16; ++s) {
    v2f v = Xp[2 * s + h];
    a[s] = v;
    pn = fmaf(v.x, v.x, fmaf(v.y, v.y, pn));
  }
  sP[wave][lane] = pn;

  v8f acc0 = {0.f, 0.f, 0.f, 0.f, 0.f, 0.f, 0.f, 0.f};
  v8f acc1 = acc0;
  const v2f* __restrict__ C0 = (const v2f*)(sC + (size_t)l16 * FDIM);
  const v2f* __restrict__ C1 = (const v2f*)(sC + (size_t)(16 + l16) * FDIM);
#pragma unroll
  for (int s = 0; s < 16; ++s) {
    v2f b0 = C0[2 * s + h];
    v2f b1 = C1[2 * s + h];
    acc0 = wmma_f32_16x16x4(a[s], b0, acc0);  // centroids 0..15
    acc1 = wmma_f32_16x16x4(a[s], b1, acc1);  // centroids 16..31
  }
  __syncthreads();  // make sP visible

  float* __restrict__ out = dist + ((size_t)b * NPTS + row0) * KCLUST;
#pragma unroll
  for (int i = 0; i < 8; ++i) {
    const int r = i + 8 * h;  // row within the 16-row strip
    float xn = sP[wave][r] + sP[wave][16 + r];
    float d0 = sqrtf(fmaxf(xn + sCn[l16]      - 2.f * acc0[i], 0.f));
    float d1 = sqrtf(fmaxf(xn + sCn[16 + l16] - 2.f * acc1[i], 0.f));
    out[(size_t)r * KCLUST + l16]      = d0;   // coalesced 64B per row
    out[(size_t)r * KCLUST + 16 + l16] = d1;
    sD[wave][r * KCLUST + l16]      = d0;
    sD[wave][r * KCLUST + 16 + l16] = d1;
  }
  __syncthreads();

  if (lane < 16) {  // lanes 0..15 each argmin one row (first-min tie-break)
    const float* drow = &sD[wave][lane * KCLUST];
    float bestv = drow[0];
    int bestk = 0;
#pragma unroll
    for (int k = 1; k < KCLUST; ++k) {
      float v = drow[k];
      if (v < bestv) { bestv = v; bestk = k; }
    }
    ids[(size_t)b * NPTS + row0 + lane] = bestk;
  }
}

extern "C" void kernel_launch(void* const* d_in, const int* in_sizes, int n_in,
                              void* d_out, int out_size, void* d_ws, size_t ws_size,
                              hipStream_t stream) {
  (void)in_sizes; (void)n_in; (void)out_size; (void)ws_size;
  const float* data = (const float*)d_in[0];

  // Workspace: cent (B*K*F f32 = 64 KB) | min_d (B*N f32 = 2 MB)
  //            | partV (B*64 f32) | partI (B*64 i32)
  char*  ws    = (char*)d_ws;
  float* cent  = (float*)ws;
  float* min_d = (float*)(ws + (size_t)BATCH * KCLUST * FDIM * sizeof(float));
  char*  ws2   = ws + (size_t)BATCH * KCLUST * FDIM * sizeof(float)
                    + (size_t)BATCH * NPTS * sizeof(float);
  float* partV = (float*)ws2;
  int*   partI = (int*)(ws2 + (size_t)BATCH * SEED_BLOCKS * sizeof(float));

  // Output: cluster_ids (B*N int32 bit patterns) then distance (B*N*K f32).
  int*   ids  = (int*)d_out;
  float* dist = (float*)d_out + (size_t)BATCH * NPTS;

  seed_first_kernel<<<BATCH, FDIM, 0, stream>>>(data, cent);
  for (int k = 1; k < KCLUST; ++k) {
    seed_update_kernel<<<dim3(SEED_BLOCKS, BATCH), 256, 0, stream>>>(
        data, cent, min_d, partV, partI, k, (k == 1) ? 1 : 0);
    seed_select_kernel<<<BATCH, SEED_BLOCKS, 0, stream>>>(
        data, partV, partI, cent, k);
  }
  dim3 grid(NPTS / 128, BATCH);
  kmeans_dist_kernel<<<grid, 256, 0, stream>>>(data, cent, dist, ids);
}